// Multiclass_Focal_Loss_32615981646280
// MI455X (gfx1250) — compile-verified
//
#include <hip/hip_runtime.h>
#include <stdint.h>

typedef float vfloat4 __attribute__((ext_vector_type(4)));
typedef int   v4i     __attribute__((ext_vector_type(4)));

#define TPB 256

// ---------------- CDNA5 async global->LDS helpers ----------------
typedef __attribute__((address_space(1))) v4i v4i_glob;
typedef __attribute__((address_space(3))) v4i v4i_lds;

__device__ __forceinline__ void async_copy_b128(void* lds_dst, const void* gsrc) {
#if __has_builtin(__builtin_amdgcn_global_load_async_to_lds_b128)
  void* g = const_cast<void*>(gsrc);
  __builtin_amdgcn_global_load_async_to_lds_b128((v4i_glob*)g, (v4i_lds*)lds_dst, 0, 0);
#else
  unsigned off = (unsigned)(unsigned long long)lds_dst;  // low 32 bits = LDS offset
  asm volatile("global_load_async_to_lds_b128 %0, %1, off"
               :: "v"(off), "v"(gsrc) : "memory");
#endif
}
__device__ __forceinline__ void wait_async_le1() {
#if __has_builtin(__builtin_amdgcn_s_wait_asynccnt)
  __builtin_amdgcn_s_wait_asynccnt(1);
#else
  asm volatile("s_wait_asynccnt 1" ::: "memory");
#endif
  asm volatile("" ::: "memory");
}
__device__ __forceinline__ void wait_async_le0() {
#if __has_builtin(__builtin_amdgcn_s_wait_asynccnt)
  __builtin_amdgcn_s_wait_asynccnt(0);
#else
  asm volatile("s_wait_asynccnt 0" ::: "memory");
#endif
  asm volatile("" ::: "memory");
}

// ---------------- K0: zero the small state region ----------------
__global__ void k_zero(unsigned* __restrict__ p, int n) {
  for (int i = blockIdx.x * blockDim.x + threadIdx.x; i < n; i += gridDim.x * blockDim.x)
    p[i] = 0u;
}

// ---------------- K1: costs + pos reduction + level-0 histogram ----------------
// scal layout: [0]=pos_cnt(u32) [1]=pos_sum(f32) [2]=k_rem [3]=prefix [4]=neg_partial(f32)
__global__ void k_cost_hist(const vfloat4* __restrict__ out4,
                            const int* __restrict__ labels,
                            unsigned* __restrict__ cost_bits,
                            unsigned* __restrict__ h0c, float* __restrict__ h0s,
                            unsigned* __restrict__ scal, int M) {
  __shared__ unsigned hc[2048];
  __shared__ float    hs[2048];
  for (int b = threadIdx.x; b < 2048; b += TPB) { hc[b] = 0u; hs[b] = 0.0f; }
  __syncthreads();

  float lsum = 0.0f; unsigned lcnt = 0u;
  const int stride = gridDim.x * TPB;
  for (int i = blockIdx.x * TPB + threadIdx.x; i < M; i += stride) {
    vfloat4 o = __builtin_nontemporal_load(&out4[i]);   // NT: read-once, don't pollute L2
    int lab   = __builtin_nontemporal_load(&labels[i]);
    float pr  = (lab == 0) ? o.x : (lab == 1) ? o.y : (lab == 2) ? o.z : o.w;
    float cost = -logf(pr);                              // cost > 0 for p in (0,1)
    if (lab > 0) {
      lsum += cost; lcnt++;
      cost_bits[i] = 0xFFFFFFFFu;                        // sentinel: never matches a prefix
    } else {
      unsigned pat = __float_as_uint(cost);              // sign=0 -> pattern order == value order
      cost_bits[i] = pat;                                // RT store: stays hot in 192MB L2
      unsigned b = (pat >> 20) & 0x7FFu;                 // bits[30:20]
      atomicAdd(&hc[b], 1u);
      atomicAdd(&hs[b], cost);
    }
  }
  // wave32 reduction of positive-class partials
  for (int off = 16; off > 0; off >>= 1) {
    lsum += __shfl_down(lsum, off);
    lcnt += __shfl_down(lcnt, off);
  }
  __shared__ float    wsum[TPB / 32];
  __shared__ unsigned wcnt[TPB / 32];
  int lane = threadIdx.x & 31, wid = threadIdx.x >> 5;
  if (lane == 0) { wsum[wid] = lsum; wcnt[wid] = lcnt; }
  __syncthreads();
  if (threadIdx.x == 0) {
    float s = 0.0f; unsigned c = 0u;
    for (int w = 0; w < TPB / 32; ++w) { s += wsum[w]; c += wcnt[w]; }
    if (c) { atomicAdd((float*)&scal[1], s); atomicAdd(&scal[0], c); }
  }
  __syncthreads();
  for (int b = threadIdx.x; b < 2048; b += TPB) {
    if (hc[b]) { atomicAdd(&h0c[b], hc[b]); atomicAdd(&h0s[b], hs[b]); }
  }
}

// ---------------- select: cooperative top-down histogram walk ----------------
// One block. Threads aggregate per-chunk count/sum into LDS (coalesced, latency-hidden);
// thread 0 then walks 256 chunk aggregates + <=per bins of the boundary chunk.
// mode 0: init k from n_pos, level-0; mode 1: level-1; mode 2: final + result
__global__ void k_select(const unsigned* __restrict__ hc, const float* __restrict__ hs,
                         int nbins, int mode, int M,
                         unsigned* __restrict__ scal, float* __restrict__ out) {
  __shared__ unsigned ccnt[TPB];
  __shared__ float    csum[TPB];
  const int per = nbins / TPB;                           // 8 (2048 bins) or 2 (512 bins)
  {
    unsigned c = 0u; float s = 0.0f;
    int base = threadIdx.x * per;
    for (int j = 0; j < per; ++j) { c += hc[base + j]; s += hs[base + j]; }
    ccnt[threadIdx.x] = c; csum[threadIdx.x] = s;
  }
  __syncthreads();
  if (threadIdx.x != 0) return;

  unsigned n_pos = scal[0];
  long long k;
  if (mode == 0) {
    float kf = floorf(2.5f * (float)n_pos);              // exact fp32, matches jnp
    k = (long long)kf;
    long long n_neg = (long long)M - (long long)n_pos;   // isfinite mask => clamp
    if (k > n_neg) k = n_neg;
  } else {
    k = (long long)scal[2];
  }

  long long cum = 0; float sum_above = 0.0f;
  int chunk = TPB - 1;
  for (int i = TPB - 1; i >= 0; --i) {                   // LDS walk: cheap
    unsigned c = ccnt[i];
    if (cum + (long long)c >= k) { chunk = i; break; }
    cum += c; sum_above += csum[i];
  }
  int b = chunk * per;
  for (int i = chunk * per + per - 1; i >= chunk * per; --i) {  // <=8 global reads
    unsigned c = hc[i];
    if (cum + (long long)c >= k) { b = i; break; }
    cum += c; sum_above += hs[i];
  }

  if (mode == 2) {
    unsigned prefix = scal[3];
    unsigned tpat = (prefix << 9) | (unsigned)b;         // exact threshold bit pattern
    float T = __uint_as_float(tpat);
    long long r = k - cum;                               // ties at threshold (equal floats)
    float negp; { float* f = (float*)&scal[4]; negp = *f; }
    float neg = negp + sum_above + (float)r * T;
    float pos_sum; { float* f = (float*)&scal[1]; pos_sum = *f; }
    float denom = floorf(3.5f * (float)n_pos);
    out[0] = (pos_sum + neg) / denom;
  } else {
    scal[2] = (unsigned)(k - cum);
    scal[3] = (mode == 0) ? (unsigned)b : ((scal[3] << 11) | (unsigned)b);
    float* negp = (float*)&scal[4];
    *negp = (mode == 0) ? sum_above : (*negp + sum_above);
  }
}

// ---------------- refine scan: async-LDS double-buffered histogram ----------------
__global__ void k_refine(const unsigned* __restrict__ cost_bits, int M,
                         const unsigned* __restrict__ scal,
                         int match_shift, int bin_shift, unsigned bin_mask, int nbins,
                         unsigned* __restrict__ hc_out, float* __restrict__ hs_out) {
  __shared__ unsigned hc[2048];
  __shared__ float    hs[2048];
  __shared__ unsigned stage[2][TPB * 4];                 // 2 x 4KB staging tiles
  for (int b = threadIdx.x; b < nbins; b += TPB) { hc[b] = 0u; hs[b] = 0.0f; }
  __syncthreads();

  const unsigned prefix = scal[3];
  const unsigned TILE = TPB * 4u;
  const unsigned ntiles = (unsigned)M / TILE;
  unsigned* my0 = &stage[0][threadIdx.x * 4];
  unsigned* my1 = &stage[1][threadIdx.x * 4];

  unsigned t0 = blockIdx.x;
  if (t0 < ntiles)
    async_copy_b128(my0, cost_bits + (size_t)t0 * TILE + threadIdx.x * 4);

  unsigned it = 0;
  for (unsigned t = t0; t < ntiles; t += gridDim.x, ++it) {
    unsigned tn = t + gridDim.x;
    if (tn < ntiles) {                                   // prefetch next tile, then wait oldest
      async_copy_b128(((it + 1) & 1) ? my1 : my0,
                      cost_bits + (size_t)tn * TILE + threadIdx.x * 4);
      wait_async_le1();
    } else {
      wait_async_le0();
    }
    const unsigned* s = (it & 1) ? my1 : my0;
#pragma unroll
    for (int j = 0; j < 4; ++j) {
      unsigned pat = s[j];
      if ((pat >> match_shift) == prefix) {              // sentinel 0xFFFFFFFF never matches
        unsigned b = (pat >> bin_shift) & bin_mask;
        atomicAdd(&hc[b], 1u);
        atomicAdd(&hs[b], __uint_as_float(pat));
      }
    }
  }
  // tail (M not multiple of TILE)
  for (int i = (int)(ntiles * TILE) + blockIdx.x * TPB + threadIdx.x; i < M;
       i += gridDim.x * TPB) {
    unsigned pat = cost_bits[i];
    if ((pat >> match_shift) == prefix) {
      unsigned b = (pat >> bin_shift) & bin_mask;
      atomicAdd(&hc[b], 1u);
      atomicAdd(&hs[b], __uint_as_float(pat));
    }
  }
  __syncthreads();
  for (int b = threadIdx.x; b < nbins; b += TPB) {
    if (hc[b]) { atomicAdd(&hc_out[b], hc[b]); atomicAdd(&hs_out[b], hs[b]); }
  }
}

// ---------------- host launch ----------------
extern "C" void kernel_launch(void* const* d_in, const int* in_sizes, int n_in,
                              void* d_out, int out_size, void* d_ws, size_t ws_size,
                              hipStream_t stream) {
  (void)n_in; (void)out_size; (void)ws_size;
  const int M = in_sizes[1];                      // labels count (outputs is M*4)
  const vfloat4* out4 = (const vfloat4*)d_in[0];
  const int* labels   = (const int*)d_in[1];

  // ws layout: [cost_bits: M u32][h0c 2048][h0s 2048][h1c 2048][h1s 2048][h2c 512][h2s 512][scal 8]
  size_t costBytes = (((size_t)M * 4u) + 255u) & ~(size_t)255u;
  unsigned* cost_bits = (unsigned*)d_ws;
  unsigned* small = (unsigned*)((char*)d_ws + costBytes);
  unsigned* h0c = small;
  float*    h0s = (float*)(small + 2048);
  unsigned* h1c = small + 4096;
  float*    h1s = (float*)(small + 6144);
  unsigned* h2c = small + 8192;
  float*    h2s = (float*)(small + 8704);
  unsigned* scal = small + 9216;
  const int smallWords = 9224;

  k_zero<<<8, TPB, 0, stream>>>(small, smallWords);
  k_cost_hist<<<2048, TPB, 0, stream>>>(out4, labels, cost_bits, h0c, h0s, scal, M);
  k_select<<<1, TPB, 0, stream>>>(h0c, h0s, 2048, 0, M, scal, (float*)d_out);
  k_refine<<<1024, TPB, 0, stream>>>(cost_bits, M, scal, 20, 9, 0x7FFu, 2048, h1c, h1s);
  k_select<<<1, TPB, 0, stream>>>(h1c, h1s, 2048, 1, M, scal, (float*)d_out);
  k_refine<<<1024, TPB, 0, stream>>>(cost_bits, M, scal, 9, 0, 0x1FFu, 512, h2c, h2s);
  k_select<<<1, TPB, 0, stream>>>(h2c, h2s, 512, 2, M, scal, (float*)d_out);
}